// GraphSAGE_47760036331738
// MI455X (gfx1250) — compile-verified
//
#include <hip/hip_runtime.h>

typedef __attribute__((ext_vector_type(2))) float v2f;
typedef __attribute__((ext_vector_type(8))) float v8f;
typedef __attribute__((ext_vector_type(4))) unsigned int u32x4;
typedef __attribute__((ext_vector_type(8))) int i32x8;
typedef __attribute__((ext_vector_type(4))) int i32x4;

#define NN 100000
#define NE 1600000

// ---------------------------------------------------------------------------
// zero a float buffer (float4 granularity; all our sizes are /4)
__global__ void sage_zero_kernel(float4* __restrict__ p, long long n4) {
    long long i = (long long)blockIdx.x * blockDim.x + threadIdx.x;
    long long stride = (long long)gridDim.x * blockDim.x;
    float4 z = {0.f, 0.f, 0.f, 0.f};
    for (; i < n4; i += stride) p[i] = z;
}

// ---------------------------------------------------------------------------
// degree count (float, exact up to 2^24) + reciprocal finalize
__global__ void sage_deg_kernel(const int* __restrict__ dst, float* __restrict__ deg) {
    int e = blockIdx.x * blockDim.x + threadIdx.x;
    if (e < NE) atomicAdd(&deg[dst[e]], 1.0f);
}
__global__ void sage_rdeg_kernel(float* __restrict__ deg) {
    int i = blockIdx.x * blockDim.x + threadIdx.x;
    if (i < NN) deg[i] = 1.0f / fmaxf(deg[i], 1.0f);
}

// ---------------------------------------------------------------------------
// edge-parallel scatter: acc[dst] += h[src], float4 chunks.
// Consecutive threads = consecutive chunks of one edge -> coalesced gathers;
// atomics land in L2 (h and acc both fit in the 192MB L2).
__global__ void sage_scatter_kernel(const float* __restrict__ h,
                                    const int* __restrict__ src,
                                    const int* __restrict__ dst,
                                    float* __restrict__ acc,
                                    int d, int dshift) {
    long long t = (long long)blockIdx.x * blockDim.x + threadIdx.x;
    long long total = (long long)NE << dshift;
    if (t >= total) return;
    int e = (int)(t >> dshift);
    int c = (int)(t & ((1 << dshift) - 1));
    int s = src[e];
    int q = dst[e];
    const float4 v = *(const float4*)(h + (long long)s * d + c * 4);
    float* p = acc + (long long)q * d + c * 4;
    atomicAdd(p + 0, v.x);
    atomicAdd(p + 1, v.y);
    atomicAdd(p + 2, v.z);
    atomicAdd(p + 3, v.w);
}

// ---------------------------------------------------------------------------
// Fused SAGE linear: out = relu?([h | acc*rdeg] @ W^T + b)
// Workgroup = 8 waves, all sharing one 64-column weight strip staged in LDS
// (TDM tensor_load_to_lds with pad: row stride K2+4 dwords -> conflict-free
// b64 LDS fragment reads). Each wave computes a 16(M) x 64(N) tile with
// V_WMMA_F32_16X16X4_F32; the [h | h_N] concat is virtual (two K phases);
// A loads are software-pipelined one K-step ahead.
template <bool RELU>
__global__ __launch_bounds__(256) void sage_gemm_kernel(
    const float* __restrict__ h,     // [NN, d]
    const float* __restrict__ acc,   // [NN, d] (un-normalized sums)
    const float* __restrict__ rdeg,  // [NN]
    const float* __restrict__ W,     // [F, 2d]
    const float* __restrict__ bias,  // [F]
    float* __restrict__ out,         // [NN, F]
    int d, int F) {
    constexpr int NT = 4;            // 4 x 16 columns per wave
    extern __shared__ float ldsB[];  // (K2+4) dwords per column row, 64 rows

    const int lane = threadIdx.x & 31;
    const int wid = threadIdx.x >> 5;  // 0..7
    const int ngroups = F >> 6;        // F / 64
    const int mtiles = NN / 16;
    const int ng = blockIdx.x % ngroups;
    const int mt = (blockIdx.x / ngroups) * 8 + wid;
    const bool valid = (mt < mtiles);

    const int K2 = 2 * d;
    const int padshift = (K2 == 512) ? 8 : 7;  // pad 2 dwords every 256/128 dwords
    const int rowstride = K2 + 4;              // dwords; mod 64 == 4 -> conflict-free
    const int nbase = ng * 64;

    // ---- stage W[nbase:nbase+64, 0:K2] into LDS (padded layout) ----
#if __has_builtin(__builtin_amdgcn_tensor_load_to_lds) && \
    __has_builtin(__builtin_amdgcn_s_wait_tensorcnt)
    if (threadIdx.x < 32) {  // wave 0 issues the TDM descriptor
        unsigned long long ga =
            (unsigned long long)(const void*)(W + (long long)nbase * K2);
        u32x4 g0;
        g0[0] = 1u;                                   // count=1, is_restore=0
        g0[1] = (unsigned)__builtin_amdgcn_groupstaticsize();  // dyn-LDS offset
        g0[2] = (unsigned)ga;                         // global_addr[31:0]
        g0[3] = (unsigned)(ga >> 32) | (2u << 30);    // addr[56:32] | type=2
        i32x8 g1;
        g1[0] = (int)((2u << 16)                      // data_size = 4 bytes
                      | (1u << 20)                    // pad_enable
                      | ((unsigned)(padshift - 1) << 22)  // interval 256/128 dw
                      | (1u << 25));                  // pad_amount = 2 dwords
        g1[1] = (int)((unsigned)(K2 & 0xFFFF) << 16); // tensor_dim0 = K2
        g1[2] = (int)((unsigned)(F & 0xFFFF) << 16);  // tensor_dim1 = F
        g1[3] = (int)((unsigned)(K2 & 0xFFFF) << 16); // tile_dim0  = K2
        g1[4] = 64;                                   // tile_dim1  = 64 rows
        g1[5] = (int)K2;                              // tensor_dim0_stride
        g1[6] = 0;
        g1[7] = 0;
        i32x4 z4 = {0, 0, 0, 0};
        i32x8 z8 = {0, 0, 0, 0, 0, 0, 0, 0};
        __builtin_amdgcn_tensor_load_to_lds(g0, g1, z4, z4, z8, 0);
        __builtin_amdgcn_s_wait_tensorcnt(0);
    }
#else
    for (int idx = threadIdx.x; idx < 64 * K2; idx += 256) {
        int n = idx / K2, k = idx - n * K2;
        ldsB[n * rowstride + k + ((k >> padshift) << 1)] =
            W[(long long)(nbase + n) * K2 + k];
    }
#endif
    __syncthreads();

    if (valid) {
        const int m0 = mt * 16;
        // A-fragment: lane l -> row m0 + l%16, K pair at 2*(l/16)
        const int arow = m0 + (lane & 15);
        const int koff = (lane >> 4) << 1;  // 0 or 2
        const float* hrow = h + (long long)arow * d;
        const float* crow = acc + (long long)arow * d;
        const float rd = rdeg[arow];
        const int ncol = lane & 15;

        // per-tile LDS fragment base (dwords): row (t*16+ncol), K pair +koff
        int boff[NT];
#pragma unroll
        for (int t = 0; t < NT; t++) boff[t] = (t * 16 + ncol) * rowstride + koff;

        v8f c[NT];
#pragma unroll
        for (int t = 0; t < NT; t++)
            c[t] = (v8f){0.f, 0.f, 0.f, 0.f, 0.f, 0.f, 0.f, 0.f};

        // phase 1: K in [0, d) -- A from h; A pipelined one K-step ahead
        v2f a = *(const v2f*)(hrow + koff);
        for (int k = 0; k < d; k += 4) {
            const int kn = (k + 4 < d) ? (k + 4) : k;
            v2f an = *(const v2f*)(hrow + kn + koff);
            const int kk = k;  // global K' = kk (+koff inside boff)
            const int kadj = kk + (((kk + koff) >> padshift) << 1);
#pragma unroll
            for (int t = 0; t < NT; t++) {
                v2f b = *(const v2f*)(ldsB + boff[t] + kadj);
                c[t] = __builtin_amdgcn_wmma_f32_16x16x4_f32(
                    false, a, false, b, (short)0, c[t], false, false);
            }
            a = an;
        }
        // phase 2: K in [d, 2d) -- A from acc * rdeg (mean aggregation)
        a = *(const v2f*)(crow + koff);
        for (int k = 0; k < d; k += 4) {
            const int kn = (k + 4 < d) ? (k + 4) : k;
            v2f an = *(const v2f*)(crow + kn + koff);
            v2f as;
            as.x = a.x * rd;
            as.y = a.y * rd;
            const int kk = d + k;
            const int kadj = kk + (((kk + koff) >> padshift) << 1);
#pragma unroll
            for (int t = 0; t < NT; t++) {
                v2f b = *(const v2f*)(ldsB + boff[t] + kadj);
                c[t] = __builtin_amdgcn_wmma_f32_16x16x4_f32(
                    false, as, false, b, (short)0, c[t], false, false);
            }
            a = an;
        }

        // store D: lane l, vgpr r -> row m0 + r + 8*(l/16), col n
        const int rbase = m0 + ((lane >> 4) << 3);
#pragma unroll
        for (int t = 0; t < NT; t++) {
            const int n = nbase + t * 16 + ncol;
            const float bv = bias[n];
#pragma unroll
            for (int r = 0; r < 8; r++) {
                float v = c[t][r] + bv;
                if (RELU) v = fmaxf(v, 0.f);
                out[(long long)(rbase + r) * F + n] = v;
            }
        }
    }
}

// ---------------------------------------------------------------------------
static inline int ceil_div_ll(long long a, int b) { return (int)((a + b - 1) / b); }

static void run_layer(const float* h, const int* src, const int* dst,
                      const float* rdeg, float* acc,
                      const float* W, const float* bias, float* out,
                      int d, int F, bool relu, hipStream_t stream) {
    // zero accumulator
    long long accN4 = (long long)NN * d / 4;
    sage_zero_kernel<<<2048, 256, 0, stream>>>((float4*)acc, accN4);
    // scatter-add h[src] into acc[dst]
    int dshift = (d == 128) ? 5 : 6;  // d/4 = 32 or 64
    long long total = (long long)NE << dshift;
    sage_scatter_kernel<<<ceil_div_ll(total, 256), 256, 0, stream>>>(h, src, dst, acc, d, dshift);
    // fused [h | acc*rdeg] @ W^T + b (+relu): 8 waves/block share LDS-staged W strip
    const int K2 = 2 * d;
    const int ngroups = F / 64;
    const int blocks = ceil_div_ll(NN / 16, 8) * ngroups;
    const size_t shmem = (size_t)(K2 + 4) * 64 * sizeof(float);
    if (relu)
        sage_gemm_kernel<true><<<blocks, 256, shmem, stream>>>(h, acc, rdeg, W, bias, out, d, F);
    else
        sage_gemm_kernel<false><<<blocks, 256, shmem, stream>>>(h, acc, rdeg, W, bias, out, d, F);
}

extern "C" void kernel_launch(void* const* d_in, const int* in_sizes, int n_in,
                              void* d_out, int out_size, void* d_ws, size_t ws_size,
                              hipStream_t stream) {
    const float* x   = (const float*)d_in[0];
    const int*   src = (const int*)d_in[1];
    const int*   dst = (const int*)d_in[2];
    const float* W1  = (const float*)d_in[3];
    const float* b1  = (const float*)d_in[4];
    const float* W2  = (const float*)d_in[5];
    const float* b2  = (const float*)d_in[6];
    const float* W3  = (const float*)d_in[7];
    const float* b3  = (const float*)d_in[8];
    float* out = (float*)d_out;

    // workspace layout (floats): rdeg[NN] | acc[NN*256] | h1[NN*256] | h2[NN*256]
    float* rdeg = (float*)d_ws;
    float* acc  = rdeg + NN;
    float* h1   = acc + (long long)NN * 256;
    float* h2   = h1 + (long long)NN * 256;

    // degree -> 1/max(deg,1), computed once (same for all layers)
    sage_zero_kernel<<<256, 256, 0, stream>>>((float4*)rdeg, NN / 4);
    sage_deg_kernel<<<ceil_div_ll(NE, 256), 256, 0, stream>>>(dst, rdeg);
    sage_rdeg_kernel<<<ceil_div_ll(NN, 256), 256, 0, stream>>>(rdeg);

    run_layer(x,  src, dst, rdeg, acc, W1, b1, h1,  128, 256, true,  stream);
    run_layer(h1, src, dst, rdeg, acc, W2, b2, h2,  256, 256, true,  stream);
    run_layer(h2, src, dst, rdeg, acc, W3, b3, out, 256, 64,  false, stream);
}